// _DummyModel_57277683860081
// MI455X (gfx1250) — compile-verified
//
#include <hip/hip_runtime.h>

// Problem constants (from reference)
#define D_FEAT   128
#define N_GRAPHS 4096

typedef __attribute__((address_space(1))) int   global_int_t;
typedef __attribute__((address_space(3))) int   lds_int_t;
typedef __attribute__((address_space(3))) float lds_float_t;

typedef unsigned int       u32;
typedef unsigned long long u64;
typedef u32 u32x4 __attribute__((ext_vector_type(4)));
typedef int i32x4 __attribute__((ext_vector_type(4)));
typedef int i32x8 __attribute__((ext_vector_type(8)));

// ---------------------------------------------------------------------------
// Zero the workspace accumulators (d_ws is poisoned to 0xAA by the harness and
// never re-poisoned between replays, so this must run every call).
// ---------------------------------------------------------------------------
__global__ void zero_ws_kernel(float* __restrict__ ws, int n) {
    int i = blockIdx.x * blockDim.x + threadIdx.x;
    if (i < n) ws[i] = 0.0f;
}

// ---------------------------------------------------------------------------
// Per-graph mean of x[:,0] over sorted `batch`.
//   - The x column is a 512B-stride gather (1M cachelines of irreducible HBM
//     traffic). One TDM descriptor per block (tile 1x256, dim0_stride=128)
//     gathers the block's column slice into LDS, offloading the strided
//     address walk to the per-WGP Tensor Data Mover (TENSORcnt).
//   - batch indices stream through LDS on the async-to-LDS path (ASYNCcnt).
//   - Wave32 segmented scan + ballot segmentation: only segment-tail lanes
//     issue global f32 atomics (~2-4 atomic pairs per wave instead of 64).
// ---------------------------------------------------------------------------
__global__ __launch_bounds__(256) void seg_mean_reduce_kernel(
    const float* __restrict__ x,
    const int*   __restrict__ batch,
    float*       __restrict__ sums,
    float*       __restrict__ counts,
    int n)
{
    const int i         = blockIdx.x * 256 + threadIdx.x;
    const unsigned lane = threadIdx.x & 31u;
    const bool active   = (i < n);

    int   seg = 0x7FFFFFFF;   // sentinel: inactive lanes form their own segment
    float v   = 0.0f;

#if defined(__gfx1250__) && __has_builtin(__builtin_amdgcn_tensor_load_to_lds) && \
    __has_builtin(__builtin_amdgcn_global_load_async_to_lds_b32)
    __shared__ float sx[256];
    __shared__ int   sbatch[256];

    const int start = blockIdx.x * 256;
    const int rem   = n - start;          // > 0 by grid construction

    if (threadIdx.x < 32) {
        // ---- Tensor DMA descriptor: gather column 0 of a 256-row slice ----
        const u64 ga       = (u64)(const char*)(x + (u64)start * D_FEAT);
        const u32 lds_base = (u32)(u64)(lds_float_t*)sx;   // LDS byte offset

        // D# group 0 (ISA 8.3): count=1, lds_addr, global_addr[56:0], type=2
        u32x4 g0;
        g0.x = 1u;
        g0.y = lds_base;
        g0.z = (u32)ga;
        g0.w = ((u32)(ga >> 32) & 0x01FFFFFFu) | 0x80000000u;   // type=2 @[127:126]

        // D# group 1 (ISA 8.4): data_size=2 (4B), tensor_dim0=128 (row len),
        // tensor_dim1=rem rows (OOB rows -> zeros), tile=1 x 256,
        // tensor_dim0_stride=128 elements (=512B row pitch).
        const u32 td0 = (u32)D_FEAT;
        const u32 td1 = (u32)rem;
        const u32 tile0 = 1u, tile1 = 256u;
        const u64 s0  = (u64)D_FEAT;
        i32x8 g1;
        g1[0] = (int)0x00020000u;                                    // data_size=4B
        g1[1] = (int)((td0 & 0xFFFFu) << 16);                        // tensor_dim0 lo
        g1[2] = (int)(((td0 >> 16) & 0xFFFFu) | ((td1 & 0xFFFFu) << 16));
        g1[3] = (int)(((td1 >> 16) & 0xFFFFu) | (tile0 << 16));      // tile_dim0
        g1[4] = (int)(tile1 & 0xFFFFu);                              // tile_dim1, dim2=0
        g1[5] = (int)(u32)(s0 & 0xFFFFFFFFu);                        // dim0_stride lo
        g1[6] = (int)(u32)((s0 >> 32) & 0xFFFFu);                    // dim0_stride hi
        g1[7] = 0;                                                   // dim1_stride=0
        i32x4 gz = {0, 0, 0, 0};                                     // 2D: groups 2/3 unused
#if __clang_major__ >= 23
        i32x8 gz8 = {0, 0, 0, 0, 0, 0, 0, 0};
        __builtin_amdgcn_tensor_load_to_lds(g0, g1, gz, gz, gz8, 0);
#else
        __builtin_amdgcn_tensor_load_to_lds(g0, g1, gz, gz, 0);
#endif
        __builtin_amdgcn_s_wait_tensorcnt(0);
    }

    // ---- batch indices: per-lane async copy to LDS (ASYNCcnt) ----
    if (active) {
        __builtin_amdgcn_global_load_async_to_lds_b32(
            (global_int_t*)(batch + i),
            (lds_int_t*)&sbatch[threadIdx.x],
            /*offset=*/0, /*cpol=*/0);
    }
    __builtin_amdgcn_s_wait_asynccnt(0);
    __syncthreads();              // publish wave0's TDM tile to all waves

    if (active) {
        v   = sx[threadIdx.x];
        seg = sbatch[threadIdx.x];
    }
#else
    if (active) {
        v   = __builtin_nontemporal_load(&x[(size_t)i * D_FEAT]);
        seg = __builtin_nontemporal_load(&batch[i]);
    }
#endif

    // ---- wave32 segmentation via ballot ----
    const int  segUp    = __shfl_up(seg, 1, 32);
    const bool headFlag = (lane == 0) || (seg != segUp);
    const unsigned hb   = (unsigned)__ballot(headFlag);   // wave32: bits [31:0]

    // Segmented inclusive scan of v (Hillis-Steele, width 32).
    bool f = headFlag;
#pragma unroll
    for (int d = 1; d < 32; d <<= 1) {
        const float v2 = __shfl_up(v, d, 32);
        const int   f2 = __shfl_up((int)f, d, 32);
        if (lane >= (unsigned)d && !f) {
            v += v2;
            f  = (f2 != 0);
        }
    }

    // Tail lane of each segment flushes sum and count (count from ballot).
    const bool tail = (lane == 31u) || ((hb >> (lane + 1u)) & 1u);
    if (active && tail) {
        const unsigned below   = hb & (0xFFFFFFFFu >> (31u - lane)); // heads at <= lane
        const unsigned headPos = 31u - (unsigned)__clz(below);
        const float    cnt     = (float)(lane - headPos + 1u);
        atomicAdd(&sums[seg],   v);     // no-return -> global_atomic_add_f32
        atomicAdd(&counts[seg], cnt);
    }
}

// ---------------------------------------------------------------------------
// out[g] = sums[g] / counts[g]   (matches reference incl. 0/0 -> NaN)
// ---------------------------------------------------------------------------
__global__ void finalize_kernel(const float* __restrict__ sums,
                                const float* __restrict__ counts,
                                float* __restrict__ out, int ngraphs) {
    int g = blockIdx.x * blockDim.x + threadIdx.x;
    if (g < ngraphs) out[g] = sums[g] / counts[g];
}

// ---------------------------------------------------------------------------
// Inputs (setup_inputs order): 0:x f32[1M,128]  1:edge_index  2:edge_attr
//                              3:batch int[1M]   (edge_* are unused)
// Output: f32[4096] (per-graph mean of column 0)
// ---------------------------------------------------------------------------
extern "C" void kernel_launch(void* const* d_in, const int* in_sizes, int n_in,
                              void* d_out, int out_size, void* d_ws, size_t ws_size,
                              hipStream_t stream) {
    const float* x     = (const float*)d_in[0];
    const int*   batch = (const int*)d_in[3];
    const int n = in_sizes[3];          // N_NODES

    float* sums   = (float*)d_ws;       // [N_GRAPHS]
    float* counts = sums + N_GRAPHS;    // [N_GRAPHS]

    zero_ws_kernel<<<(2 * N_GRAPHS + 255) / 256, 256, 0, stream>>>(sums, 2 * N_GRAPHS);
    seg_mean_reduce_kernel<<<(n + 255) / 256, 256, 0, stream>>>(x, batch, sums, counts, n);
    finalize_kernel<<<(N_GRAPHS + 255) / 256, 256, 0, stream>>>(sums, counts, (float*)d_out, N_GRAPHS);
}